// TokenMambaStack_41635412967477
// MI455X (gfx1250) — compile-verified
//
#include <hip/hip_runtime.h>
#include <hip/hip_bf16.h>
#include <math.h>

// ---------------------------------------------------------------------------
// Mamba stack (2 layers) for MI455X / gfx1250, wave32, WMMA f32 path.
// ---------------------------------------------------------------------------

#define D_MODEL 256
#define DEPTH   2
#define D_INNER 512
#define D_STATE 16
#define D_CONV  4
#define DT_RANK 16
#define BATCH   2
#define SEQLEN  2048
#define NROWS   (BATCH * SEQLEN)        // 4096 token rows
#define DBL_N   (DT_RANK + 2 * D_STATE) // 48

typedef __attribute__((ext_vector_type(2))) float v2f;
typedef __attribute__((ext_vector_type(4))) float v4f;
typedef __attribute__((ext_vector_type(8))) float v8f;

// ---------------------------------------------------------------------------
// LayerNorm: one wave per token row (256 elems -> 8 per lane).
// ---------------------------------------------------------------------------
__global__ __launch_bounds__(32)
void ln_kernel(const float* __restrict__ x,
               const float* __restrict__ w,
               const float* __restrict__ b,
               float* __restrict__ out)
{
    const int row  = blockIdx.x;
    const int lane = threadIdx.x;
    const float* rp = x + (size_t)row * D_MODEL + lane * 8;

    v4f a = *(const v4f*)(rp);
    v4f c = *(const v4f*)(rp + 4);

    float s  = a[0]+a[1]+a[2]+a[3] + c[0]+c[1]+c[2]+c[3];
    float sq = a[0]*a[0]+a[1]*a[1]+a[2]*a[2]+a[3]*a[3]
             + c[0]*c[0]+c[1]*c[1]+c[2]*c[2]+c[3]*c[3];
    #pragma unroll
    for (int off = 16; off >= 1; off >>= 1) {
        s  += __shfl_xor(s,  off);
        sq += __shfl_xor(sq, off);
    }
    const float mu   = s * (1.0f / D_MODEL);
    const float var  = sq * (1.0f / D_MODEL) - mu * mu;
    const float rstd = rsqrtf(var + 1e-5f);

    float* op = out + (size_t)row * D_MODEL + lane * 8;
    #pragma unroll
    for (int j = 0; j < 8; ++j) {
        const int cidx = lane * 8 + j;
        const float v = (j < 4 ? a[j] : c[j - 4]);
        op[j] = (v - mu) * rstd * w[cidx] + b[cidx];
    }
}

// ---------------------------------------------------------------------------
// WMMA FP32 GEMM, fully compile-time shapes, N-register-blocked:
//   C[m,n] = act( sum_k A[m,k]*W[n,k] + bias[n] ) + res[m,n]
// One wave computes a 16 x (16*NSUB) tile. The K loop is staged 4 K-steps
// deep: all 4*(NSUB+1) fragment loads are issued as a cluster (constant
// immediate offsets, since LDA/LDW/K are template constants), then the 4*NSUB
// WMMAs drain them with staggered s_wait_loadcnt instead of load->wmma
// ping-pong. V_WMMA_F32_16X16X4_F32 layouts:
//   lane L: A[M=L%16][K=k+2*(L/16)+v], B[K=k+2*(L/16)+v][N=L%16],
//   D VGPR v -> row (v + 8*(L/16)), col (L%16).
// ACT: 0 = none, 1 = softplus
// ---------------------------------------------------------------------------
template<int NSUB, int LDA, int LDW, int K, int ACT>
__global__ __launch_bounds__(32)
void wmma_gemm_f32(const float* __restrict__ A,
                   const float* __restrict__ W,
                   const float* __restrict__ bias,
                   const float* __restrict__ res, int ldres,
                   float* __restrict__ C, int ldc)
{
    constexpr int KSTEP  = 4;                       // K per WMMA
    constexpr int STAGE  = (K >= 16) ? 4 : 1;       // K-steps staged per window
    constexpr int KWIN   = KSTEP * STAGE;           // K per window

    const int tileN0 = blockIdx.x * NSUB;
    const int tileM  = blockIdx.y;
    const int lane   = threadIdx.x;
    const int half   = lane >> 4;   // 0: K pair {0,1}; 1: K pair {2,3}
    const int l16    = lane & 15;

    const float* arow = A + (size_t)(tileM * 16 + l16) * LDA + 2 * half;
    const float* wrow[NSUB];
    #pragma unroll
    for (int t = 0; t < NSUB; ++t)
        wrow[t] = W + (size_t)((tileN0 + t) * 16 + l16) * LDW + 2 * half;

    v8f acc[NSUB];
    #pragma unroll
    for (int t = 0; t < NSUB; ++t) acc[t] = (v8f){};

    for (int k = 0; k < K; k += KWIN) {
        // ---- stage all fragment loads for this window (clustered) ----
        v2f a[STAGE];
        v2f b[STAGE][NSUB];
        #pragma unroll
        for (int u = 0; u < STAGE; ++u)
            a[u] = *(const v2f*)(arow + k + KSTEP * u);
        #pragma unroll
        for (int u = 0; u < STAGE; ++u)
            #pragma unroll
            for (int t = 0; t < NSUB; ++t)
                b[u][t] = *(const v2f*)(wrow[t] + k + KSTEP * u);

        // ---- drain: 4*NSUB WMMAs against the staged fragments ----
        #pragma unroll
        for (int u = 0; u < STAGE; ++u)
            #pragma unroll
            for (int t = 0; t < NSUB; ++t)
                acc[t] = __builtin_amdgcn_wmma_f32_16x16x4_f32(
                    /*neg_a=*/false, a[u], /*neg_b=*/false, b[u][t],
                    /*c_mod=*/(short)0, acc[t],
                    /*reuse_a=*/false, /*reuse_b=*/false);
    }

    #pragma unroll
    for (int t = 0; t < NSUB; ++t) {
        const int col = (tileN0 + t) * 16 + l16;
        const float bv = bias ? bias[col] : 0.0f;
        #pragma unroll
        for (int v = 0; v < 8; ++v) {
            const int row = tileM * 16 + v + 8 * half;
            float val = acc[t][v] + bv;
            if (ACT == 1) {                       // softplus
                val = (val > 20.0f) ? val : log1pf(__expf(val));
            }
            if (res) val += res[(size_t)row * ldres + col];
            C[(size_t)row * ldc + col] = val;
        }
    }
}

// ---------------------------------------------------------------------------
// Depthwise causal conv (width 4) + SiLU over the xi half of xz.
// ---------------------------------------------------------------------------
__global__ __launch_bounds__(256)
void conv_silu_kernel(const float* __restrict__ xz,
                      const float* __restrict__ cw,   // [D_INNER][D_CONV]
                      const float* __restrict__ cb,   // [D_INNER]
                      float* __restrict__ xc)         // [NROWS][D_INNER]
{
    const int t = blockIdx.x * blockDim.x + threadIdx.x;
    if (t >= NROWS * D_INNER) return;
    const int d = t % D_INNER;
    const int l = (t / D_INNER) % SEQLEN;
    const int b = t / (D_INNER * SEQLEN);

    float acc = cb[d];
    #pragma unroll
    for (int j = 0; j < D_CONV; ++j) {
        const int ls = l - (D_CONV - 1) + j;
        if (ls >= 0) {
            acc += cw[d * D_CONV + j] *
                   xz[((size_t)(b * SEQLEN + ls)) * (2 * D_INNER) + d];
        }
    }
    const float s = acc / (1.0f + __expf(-acc));  // silu
    xc[(size_t)t] = s;
}

// ---------------------------------------------------------------------------
// Selective scan. One lane per (batch, channel, state); 16 lanes per channel,
// cross-lane y-reduction via shfl_xor (xor masks < 16 never cross 16-lane
// halves of a wave32). Software-pipelined one step deep + prefetch 8 rows
// ahead (global_prefetch_b8) so loads overlap the h-recurrence chain.
// Fused epilogue: y = (scan + u*D) * silu(z).
// ---------------------------------------------------------------------------
__global__ __launch_bounds__(256)
void scan_kernel(const float* __restrict__ delta, // [NROWS][D_INNER]
                 const float* __restrict__ xc,    // [NROWS][D_INNER]
                 const float* __restrict__ dbl,   // [NROWS][48]
                 const float* __restrict__ xz,    // [NROWS][2*D_INNER] (z half)
                 const float* __restrict__ alog,  // [D_INNER][D_STATE]
                 const float* __restrict__ dpar,  // [D_INNER]
                 float* __restrict__ yf)          // [NROWS][D_INNER]
{
    const int n = threadIdx.x & 15;
    const int c = blockIdx.x * (blockDim.x >> 4) + (threadIdx.x >> 4);
    const int b = c >> 9;        // / D_INNER
    const int d = c & 511;       // % D_INNER

    const float Aval = -__expf(alog[d * D_STATE + n]);
    const float Dval = dpar[d];

    const size_t r0 = (size_t)b * SEQLEN;
    const float* pD = delta + r0 * D_INNER + d;
    const float* pU = xc    + r0 * D_INNER + d;
    const float* pB = dbl   + r0 * DBL_N + DT_RANK + n;
    const float* pC = dbl   + r0 * DBL_N + DT_RANK + D_STATE + n;
    const float* pZ = xz    + r0 * (2 * D_INNER) + D_INNER + d;
    float*       pY = yf    + r0 * D_INNER + d;

    const int PF = 8;  // prefetch distance (rows ahead)

    // pipeline prologue: load step 0
    float dlt = pD[0];
    float u   = pU[0];
    float Bn  = pB[0];
    float Cn  = pC[0];

    float h = 0.0f;
    for (int l = 0; l < SEQLEN; ++l) {
        // issue next step's loads before the dependent math
        float dlt_n = 0.f, u_n = 0.f, Bn_n = 0.f, Cn_n = 0.f;
        if (l + 1 < SEQLEN) {
            dlt_n = pD[(size_t)(l + 1) * D_INNER];
            u_n   = pU[(size_t)(l + 1) * D_INNER];
            Bn_n  = pB[(size_t)(l + 1) * DBL_N];
            Cn_n  = pC[(size_t)(l + 1) * DBL_N];
        }
        if (l + PF < SEQLEN) {
            __builtin_prefetch(pD + (size_t)(l + PF) * D_INNER, 0, 1);
            __builtin_prefetch(pB + (size_t)(l + PF) * DBL_N,   0, 1);
        }

        const float dA = __expf(dlt * Aval);
        h = dA * h + dlt * Bn * u;

        float y = h * Cn;
        y += __shfl_xor(y, 1);
        y += __shfl_xor(y, 2);
        y += __shfl_xor(y, 4);
        y += __shfl_xor(y, 8);

        if (n == 0) {
            const float z  = pZ[(size_t)l * (2 * D_INNER)];
            const float sz = z / (1.0f + __expf(-z));
            pY[(size_t)l * D_INNER] = (y + u * Dval) * sz;
        }

        dlt = dlt_n; u = u_n; Bn = Bn_n; Cn = Cn_n;
    }
}

// ---------------------------------------------------------------------------
// Host-side pipeline
// ---------------------------------------------------------------------------
extern "C" void kernel_launch(void* const* d_in, const int* in_sizes, int n_in,
                              void* d_out, int out_size, void* d_ws, size_t ws_size,
                              hipStream_t stream)
{
    const float* x     = (const float*)d_in[0];
    const float* ln_w  = (const float*)d_in[1];
    const float* ln_b  = (const float*)d_in[2];
    const float* ipw   = (const float*)d_in[3];
    const float* cw    = (const float*)d_in[4];
    const float* cb    = (const float*)d_in[5];
    const float* xpw   = (const float*)d_in[6];
    const float* dpw   = (const float*)d_in[7];
    const float* dpb   = (const float*)d_in[8];
    const float* alog  = (const float*)d_in[9];
    const float* dpar  = (const float*)d_in[10];
    const float* opw   = (const float*)d_in[11];
    float* out = (float*)d_out;

    // workspace layout (floats)
    float* ws    = (float*)d_ws;
    float* hnorm = ws;                                 // NROWS * D_MODEL
    float* xz    = hnorm + (size_t)NROWS * D_MODEL;    // NROWS * 2*D_INNER
    float* xc    = xz    + (size_t)NROWS * 2 * D_INNER;// NROWS * D_INNER
    float* dbl   = xc    + (size_t)NROWS * D_INNER;    // NROWS * 48
    float* delta = dbl   + (size_t)NROWS * DBL_N;      // NROWS * D_INNER
    float* yf    = delta + (size_t)NROWS * D_INNER;    // NROWS * D_INNER
    float* x1    = yf    + (size_t)NROWS * D_INNER;    // NROWS * D_MODEL

    for (int i = 0; i < DEPTH; ++i) {
        const float* xin  = (i == 0) ? x : x1;
        float*       xout = (i == DEPTH - 1) ? out : x1;

        // 1. LayerNorm
        ln_kernel<<<NROWS, 32, 0, stream>>>(
            xin, ln_w + i * D_MODEL, ln_b + i * D_MODEL, hnorm);

        // 2. in_proj: [4096,256] x [1024,256]^T -> [4096,1024]   (16x64/wave)
        wmma_gemm_f32<4, D_MODEL, D_MODEL, D_MODEL, 0>
            <<<dim3(2 * D_INNER / 64, NROWS / 16), 32, 0, stream>>>(
            hnorm,
            ipw + (size_t)i * 2 * D_INNER * D_MODEL,
            nullptr, nullptr, 0,
            xz, 2 * D_INNER);

        // 3. depthwise conv + silu
        conv_silu_kernel<<<(NROWS * D_INNER) / 256, 256, 0, stream>>>(
            xz, cw + (size_t)i * D_INNER * D_CONV, cb + i * D_INNER, xc);

        // 4. x_proj: [4096,512] x [48,512]^T -> [4096,48]        (16x48/wave)
        wmma_gemm_f32<3, D_INNER, D_INNER, D_INNER, 0>
            <<<dim3(DBL_N / 48, NROWS / 16), 32, 0, stream>>>(
            xc,
            xpw + (size_t)i * DBL_N * D_INNER,
            nullptr, nullptr, 0,
            dbl, DBL_N);

        // 5. dt_proj + softplus: [4096,16] x [512,16]^T -> [4096,512]
        wmma_gemm_f32<4, DBL_N, DT_RANK, DT_RANK, 1>
            <<<dim3(D_INNER / 64, NROWS / 16), 32, 0, stream>>>(
            dbl,
            dpw + (size_t)i * D_INNER * DT_RANK,
            dpb + i * D_INNER, nullptr, 0,
            delta, D_INNER);

        // 6. selective scan (sequential over L, parallel over B*D_INNER*D_STATE)
        scan_kernel<<<(BATCH * D_INNER * D_STATE) / 256, 256, 0, stream>>>(
            delta, xc, dbl, xz,
            alog + (size_t)i * D_INNER * D_STATE, dpar + i * D_INNER, yf);

        // 7. out_proj + residual: [4096,512] x [256,512]^T + xin -> [4096,256]
        wmma_gemm_f32<4, D_INNER, D_INNER, D_INNER, 0>
            <<<dim3(D_MODEL / 64, NROWS / 16), 32, 0, stream>>>(
            yf,
            opw + (size_t)i * D_MODEL * D_INNER,
            nullptr, xin, D_MODEL,
            xout, D_MODEL);
    }
}